// TwoMaxLayerPoolingAggregator_89146341196170
// MI455X (gfx1250) — compile-verified
//
#include <hip/hip_runtime.h>

typedef __attribute__((ext_vector_type(2))) float v2f;
typedef __attribute__((ext_vector_type(4))) float v4f;
typedef __attribute__((ext_vector_type(8))) float v8f;

#define NEG_INF_F (-1e10f)
#define DIM 128      // feature dim (in == out)
#define KN 32        // max neighbors
#define ROWS 32      // rows (nodes) per block
#define LDSS 132     // LDS row stride in floats: bank = (4r + k) % 64, conflict-free

__global__ __launch_bounds__(256) void
TwoMaxLayerPoolingAggregator_kernel(const float* __restrict__ self_vecs,
                                    const float* __restrict__ neigh_vecs,
                                    const int*   __restrict__ adj,
                                    const float* __restrict__ W,
                                    float*       __restrict__ out,
                                    int N)
{
    __shared__ float Ah[ROWS * LDSS];   // max-pooled neigh_h tile, 32x128 (padded)

    const int tid      = threadIdx.x;
    const int lane     = tid & 31;
    const int wv       = tid >> 5;                 // wave id 0..7
    const int tileBase = blockIdx.x * ROWS;

    // ---------------- Phase 1: masked neighbor max-pool -> LDS, self+ReLU -> out
    // Mapping: one wave per row per pass; lane owns 4 consecutive features (b128 loads).
    const int d4 = lane * 4;
    for (int pass = 0; pass < ROWS / 8; ++pass) {
        const int r        = wv + pass * 8;
        const int row      = tileBase + r;
        const bool rowValid = (row < N);

        // Build per-row neighbor validity bitmask with wave32 ballot:
        // lane k supplies bit k for this row's neighbor k.
        int aval = rowValid ? adj[(size_t)row * KN + lane] : 0;
        unsigned mask = (unsigned)__ballot(aval != 0);

        float init = rowValid ? NEG_INF_F : 0.0f;
        v4f m4; m4.x = init; m4.y = init; m4.z = init; m4.w = init;

        if (rowValid) {
            const float* base = neigh_vecs + ((size_t)row * KN) * DIM + d4;
            #pragma unroll 4
            for (int k = 0; k < KN; ++k) {
                // wave-uniform predicate: whole wave skips the 512B load when
                // the neighbor slot is invalid (~halves HBM traffic)
                if ((mask >> k) & 1u) {
                    v4f v = __builtin_nontemporal_load((const v4f*)(base + (size_t)k * DIM));
                    m4.x = fmaxf(m4.x, v.x);
                    m4.y = fmaxf(m4.y, v.y);
                    m4.z = fmaxf(m4.z, v.z);
                    m4.w = fmaxf(m4.w, v.w);
                }
            }
            // self half of the output: relu(self_vecs) -> out[row, 0:128]
            v4f s = __builtin_nontemporal_load((const v4f*)(self_vecs + (size_t)row * DIM + d4));
            s.x = fmaxf(s.x, 0.0f); s.y = fmaxf(s.y, 0.0f);
            s.z = fmaxf(s.z, 0.0f); s.w = fmaxf(s.w, 0.0f);
            __builtin_nontemporal_store(s, (v4f*)(out + (size_t)row * (2 * DIM) + d4));
        }
        *(v4f*)(&Ah[r * LDSS + d4]) = m4;   // 16B-aligned (528*r + 16*lane)
    }
    __syncthreads();

    // ---------------- Phase 2: neigh_h[32x128] @ W[128x128] via V_WMMA_F32_16X16X4_F32
    // Wave wv owns output columns n = 16*wv .. 16*wv+15; two row-tiles share each B frag.
    const int hs  = lane >> 4;       // half-wave select
    const int l15 = lane & 15;
    const int mh  = hs * 8;          // C/D row offset within 16
    const int n   = wv * 16 + l15;   // output column

    v8f acc0 = {};
    v8f acc1 = {};
    const float* Wc = W + n;         // column n of the 128x128 weight (row-major)

    #pragma unroll 8
    for (int k = 0; k < DIM; k += 4) {
        const int kb = k + hs * 2;   // K pair handled by this half-wave

        v2f b;                        // B frag: W[kb][n], W[kb+1][n]
        b.x = Wc[(size_t)kb * DIM];
        b.y = Wc[(size_t)(kb + 1) * DIM];

        v2f a0;                       // A frag, rows 0..15
        a0.x = Ah[l15 * LDSS + kb];
        a0.y = Ah[l15 * LDSS + kb + 1];
        v2f a1;                       // A frag, rows 16..31
        a1.x = Ah[(16 + l15) * LDSS + kb];
        a1.y = Ah[(16 + l15) * LDSS + kb + 1];

        // (neg_a, A, neg_b, B, c_mod, C, reuse_a, reuse_b)
        acc0 = __builtin_amdgcn_wmma_f32_16x16x4_f32(false, a0, false, b, (short)0, acc0, false, false);
        acc1 = __builtin_amdgcn_wmma_f32_16x16x4_f32(false, a1, false, b, (short)0, acc1, false, false);
    }

    // ---------------- Epilogue: relu(from_neighs) -> out[row, 128:256]
    #pragma unroll
    for (int v = 0; v < 8; ++v) {
        const int m    = mh + v;
        const int row0 = tileBase + m;
        const int row1 = tileBase + 16 + m;
        if (row0 < N)
            out[(size_t)row0 * (2 * DIM) + DIM + n] = fmaxf(acc0[v], 0.0f);
        if (row1 < N)
            out[(size_t)row1 * (2 * DIM) + DIM + n] = fmaxf(acc1[v], 0.0f);
    }
}

extern "C" void kernel_launch(void* const* d_in, const int* in_sizes, int n_in,
                              void* d_out, int out_size, void* d_ws, size_t ws_size,
                              hipStream_t stream) {
    (void)n_in; (void)out_size; (void)d_ws; (void)ws_size;
    const float* self_vecs  = (const float*)d_in[0];
    const float* neigh_vecs = (const float*)d_in[1];
    const int*   adj        = (const int*)  d_in[2];
    const float* W          = (const float*)d_in[3];
    float*       out        = (float*)d_out;

    const int N = in_sizes[0] / DIM;                 // 50000
    const int blocks = (N + ROWS - 1) / ROWS;        // 1563

    TwoMaxLayerPoolingAggregator_kernel<<<blocks, 256, 0, stream>>>(
        self_vecs, neigh_vecs, adj, W, out, N);
}